// ChannelAwareAttentionModule_40973988004242
// MI455X (gfx1250) — compile-verified
//
#include <hip/hip_runtime.h>
#include <stdint.h>

typedef __attribute__((ext_vector_type(2))) float v2f;
typedef __attribute__((ext_vector_type(8))) float v8f;

#define B_   16
#define C_   256
#define CI_  128
#define N_   4096
#define EPS_ 1e-5f

__device__ __forceinline__ v8f wmma_f32(v2f a, v2f b, v8f c) {
  // D = A(16x4 f32) * B(4x16 f32) + C(16x16 f32)
  return __builtin_amdgcn_wmma_f32_16x16x4_f32(false, a, false, b, (short)0, c, false, false);
}

// Async global->LDS 16-byte copy (CDNA5, tracked by ASYNCcnt). GV mode.
__device__ __forceinline__ void async_b128(uint32_t lds_off, uint64_t gaddr) {
  asm volatile("global_load_async_to_lds_b128 %0, %1, off"
               :: "v"(lds_off), "v"(gaddr) : "memory");
}
__device__ __forceinline__ void wait_async0() {
  asm volatile("s_wait_asynccnt 0" ::: "memory");
}
__device__ __forceinline__ void wait_async4() {
  asm volatile("s_wait_asynccnt 4" ::: "memory");
}
// Low 32 bits of a flat LDS-aperture address are the LDS byte offset.
__device__ __forceinline__ uint32_t lds_off(const void* p) {
  return (uint32_t)(uintptr_t)p;
}

// ---------------------------------------------------------------------------
// Generic NN GEMM:  C[b] = A[b](128*gridDim.y x K) @ Bm[b](K x N) (+ bias)
// Workgroup: 128x128 block, 8 waves (4x2), each wave 2x4 16x16 tiles.
// Double-buffered async global->LDS staging.
// ---------------------------------------------------------------------------
__global__ __launch_bounds__(256) void gemm_nn_wmma(
    const float* __restrict__ A, long long sA,
    const float* __restrict__ Bm, long long sB,
    float* __restrict__ Cm, long long sC,
    const float* __restrict__ bias,
    int K, int N)
{
  __shared__ float As[2][128][20];   // stride 20: 16B-aligned rows, conflict-free
  __shared__ float Bs[2][16][128];

  const int t  = threadIdx.x;
  const int w  = t >> 5, l = t & 31, lh = l >> 4, ll = l & 15;
  const int wm = w >> 1, wn = w & 1;

  const float* Ab = A  + (size_t)blockIdx.z * sA + (size_t)blockIdx.y * 128 * K;
  const float* Bb = Bm + (size_t)blockIdx.z * sB;
  float*       Cb = Cm + (size_t)blockIdx.z * sC + (size_t)blockIdx.y * 128 * N;
  const int nBase = blockIdx.x * 128;

  // per-thread async chunks: A tile = 512 x b128, B tile = 512 x b128, 2 each
  const int ca0 = t, ca1 = t + 256;
  const uint64_t aSrc0 = (uint64_t)(uintptr_t)(Ab + (size_t)(ca0 >> 2) * K + (ca0 & 3) * 4);
  const uint64_t aSrc1 = (uint64_t)(uintptr_t)(Ab + (size_t)(ca1 >> 2) * K + (ca1 & 3) * 4);
  const uint64_t bSrc0 = (uint64_t)(uintptr_t)(Bb + (size_t)(ca0 >> 5) * N + nBase + (ca0 & 31) * 4);
  const uint64_t bSrc1 = (uint64_t)(uintptr_t)(Bb + (size_t)(ca1 >> 5) * N + nBase + (ca1 & 31) * 4);
  const uint32_t aDst0 = lds_off(&As[0][ca0 >> 2][(ca0 & 3) * 4]);
  const uint32_t aDst1 = lds_off(&As[0][ca1 >> 2][(ca1 & 3) * 4]);
  const uint32_t bDst0 = lds_off(&Bs[0][ca0 >> 5][(ca0 & 31) * 4]);
  const uint32_t bDst1 = lds_off(&Bs[0][ca1 >> 5][(ca1 & 31) * 4]);
  const uint32_t aBuf  = (uint32_t)((const char*)&As[1][0][0] - (const char*)&As[0][0][0]);
  const uint32_t bBuf  = (uint32_t)((const char*)&Bs[1][0][0] - (const char*)&Bs[0][0][0]);
  const uint64_t aStepK = 64;                       // 16 floats along K
  const uint64_t bStepK = (uint64_t)16 * N * 4;     // 16 rows of B

  v8f acc[2][4];
#pragma unroll
  for (int mi = 0; mi < 2; ++mi)
#pragma unroll
    for (int ni = 0; ni < 4; ++ni)
#pragma unroll
      for (int r = 0; r < 8; ++r) acc[mi][ni][r] = 0.0f;

  const int nk = K >> 4;
  // prologue: stage chunk 0 into buffer 0
  async_b128(aDst0, aSrc0);
  async_b128(aDst1, aSrc1);
  async_b128(bDst0, bSrc0);
  async_b128(bDst1, bSrc1);

  for (int kc = 0; kc < nk; ++kc) {
    const int cur = kc & 1;
    if (kc + 1 < nk) {   // prefetch next chunk into the other buffer
      const uint32_t ab = (cur ^ 1) ? aBuf : 0u;
      const uint32_t bb = (cur ^ 1) ? bBuf : 0u;
      const uint64_t ka = (uint64_t)(kc + 1) * aStepK;
      const uint64_t kb = (uint64_t)(kc + 1) * bStepK;
      async_b128(aDst0 + ab, aSrc0 + ka);
      async_b128(aDst1 + ab, aSrc1 + ka);
      async_b128(bDst0 + bb, bSrc0 + kb);
      async_b128(bDst1 + bb, bSrc1 + kb);
      wait_async4();     // current chunk's 4 transfers complete (in-order)
    } else {
      wait_async0();
    }
    __syncthreads();

#pragma unroll
    for (int ks = 0; ks < 16; ks += 4) {
      v2f af[2], bf[4];
#pragma unroll
      for (int mi = 0; mi < 2; ++mi)
        af[mi] = *(const v2f*)&As[cur][(wm * 2 + mi) * 16 + ll][ks + 2 * lh];
#pragma unroll
      for (int ni = 0; ni < 4; ++ni) {
        const int col = (wn * 4 + ni) * 16 + ll;
        v2f b;
        b.x = Bs[cur][ks + 2 * lh + 0][col];
        b.y = Bs[cur][ks + 2 * lh + 1][col];
        bf[ni] = b;
      }
#pragma unroll
      for (int mi = 0; mi < 2; ++mi)
#pragma unroll
        for (int ni = 0; ni < 4; ++ni)
          acc[mi][ni] = wmma_f32(af[mi], bf[ni], acc[mi][ni]);
    }
    __syncthreads();
  }

#pragma unroll
  for (int mi = 0; mi < 2; ++mi)
#pragma unroll
    for (int r = 0; r < 8; ++r) {
      const int row = (wm * 2 + mi) * 16 + r + 8 * lh;
      const float bv = bias ? bias[blockIdx.y * 128 + row] : 0.0f;
#pragma unroll
      for (int ni = 0; ni < 4; ++ni) {
        const int col = nBase + (wn * 4 + ni) * 16 + ll;
        Cb[(size_t)row * N + col] = acc[mi][ni][r] + bv;
      }
    }
}

// ---------------------------------------------------------------------------
// NT Gram with split-K:  Fp[b][split] = Th[b](128xKslice) @ Ph[b](128xKslice)^T
// Kslice = 256, 16 splits covering K = 4096. Double-buffered async staging.
// ---------------------------------------------------------------------------
__global__ __launch_bounds__(256) void gemm_nt_splitk(
    const float* __restrict__ Th, const float* __restrict__ Ph,
    float* __restrict__ Fp)
{
  __shared__ float As[2][128][20];
  __shared__ float Bs[2][128][20];

  const int t  = threadIdx.x;
  const int w  = t >> 5, l = t & 31, lh = l >> 4, ll = l & 15;
  const int wm = w >> 1, wn = w & 1;

  const int b = blockIdx.z;
  const int kBase = blockIdx.x * 256;
  const float* Ab = Th + (size_t)b * ((size_t)CI_ * N_) + kBase;
  const float* Bb = Ph + (size_t)b * ((size_t)CI_ * N_) + kBase;
  float* Fb = Fp + ((size_t)b * gridDim.x + blockIdx.x) * (128 * 128);

  const int ca0 = t, ca1 = t + 256;
  const uint64_t aSrc0 = (uint64_t)(uintptr_t)(Ab + (size_t)(ca0 >> 2) * N_ + (ca0 & 3) * 4);
  const uint64_t aSrc1 = (uint64_t)(uintptr_t)(Ab + (size_t)(ca1 >> 2) * N_ + (ca1 & 3) * 4);
  const uint64_t bSrc0 = (uint64_t)(uintptr_t)(Bb + (size_t)(ca0 >> 2) * N_ + (ca0 & 3) * 4);
  const uint64_t bSrc1 = (uint64_t)(uintptr_t)(Bb + (size_t)(ca1 >> 2) * N_ + (ca1 & 3) * 4);
  const uint32_t aDst0 = lds_off(&As[0][ca0 >> 2][(ca0 & 3) * 4]);
  const uint32_t aDst1 = lds_off(&As[0][ca1 >> 2][(ca1 & 3) * 4]);
  const uint32_t bDst0 = lds_off(&Bs[0][ca0 >> 2][(ca0 & 3) * 4]);
  const uint32_t bDst1 = lds_off(&Bs[0][ca1 >> 2][(ca1 & 3) * 4]);
  const uint32_t aBuf  = (uint32_t)((const char*)&As[1][0][0] - (const char*)&As[0][0][0]);
  const uint32_t bBuf  = (uint32_t)((const char*)&Bs[1][0][0] - (const char*)&Bs[0][0][0]);

  v8f acc[2][4];
#pragma unroll
  for (int mi = 0; mi < 2; ++mi)
#pragma unroll
    for (int ni = 0; ni < 4; ++ni)
#pragma unroll
      for (int r = 0; r < 8; ++r) acc[mi][ni][r] = 0.0f;

  const int nk = 256 / 16;
  async_b128(aDst0, aSrc0);
  async_b128(aDst1, aSrc1);
  async_b128(bDst0, bSrc0);
  async_b128(bDst1, bSrc1);

  for (int kc = 0; kc < nk; ++kc) {
    const int cur = kc & 1;
    if (kc + 1 < nk) {
      const uint32_t ab = (cur ^ 1) ? aBuf : 0u;
      const uint32_t bb = (cur ^ 1) ? bBuf : 0u;
      const uint64_t ka = (uint64_t)(kc + 1) * 64;
      async_b128(aDst0 + ab, aSrc0 + ka);
      async_b128(aDst1 + ab, aSrc1 + ka);
      async_b128(bDst0 + bb, bSrc0 + ka);
      async_b128(bDst1 + bb, bSrc1 + ka);
      wait_async4();
    } else {
      wait_async0();
    }
    __syncthreads();

#pragma unroll
    for (int ks = 0; ks < 16; ks += 4) {
      v2f af[2], bf[4];
#pragma unroll
      for (int mi = 0; mi < 2; ++mi)
        af[mi] = *(const v2f*)&As[cur][(wm * 2 + mi) * 16 + ll][ks + 2 * lh];
#pragma unroll
      for (int ni = 0; ni < 4; ++ni)
        bf[ni] = *(const v2f*)&Bs[cur][(wn * 4 + ni) * 16 + ll][ks + 2 * lh];
#pragma unroll
      for (int mi = 0; mi < 2; ++mi)
#pragma unroll
        for (int ni = 0; ni < 4; ++ni)
          acc[mi][ni] = wmma_f32(af[mi], bf[ni], acc[mi][ni]);
    }
    __syncthreads();
  }

#pragma unroll
  for (int mi = 0; mi < 2; ++mi)
#pragma unroll
    for (int r = 0; r < 8; ++r) {
      const int row = (wm * 2 + mi) * 16 + r + 8 * lh;
#pragma unroll
      for (int ni = 0; ni < 4; ++ni) {
        const int col = (wn * 4 + ni) * 16 + ll;
        Fb[row * 128 + col] = acc[mi][ni][r];
      }
    }
}

// Deterministic split-K reduction: F = sum over 16 partials
__global__ __launch_bounds__(256) void reduce_split(const float* __restrict__ Fp,
                                                    float* __restrict__ F) {
  const int idx = blockIdx.x * 256 + threadIdx.x;   // B_*128*128 total
  const int b = idx >> 14;
  const int r = idx & 16383;
  float s = 0.0f;
#pragma unroll
  for (int sp = 0; sp < 16; ++sp)
    s += Fp[((size_t)b * 16 + sp) * 16384 + r];
  F[idx] = s;
}

// Softmax over rows of length 128 (with max subtraction; logits have sigma~64)
__global__ __launch_bounds__(128) void softmax_row128(float* __restrict__ F) {
  __shared__ float red[128];
  const int t = threadIdx.x;
  float* row = F + (size_t)blockIdx.x * 128;
  float v = row[t];
  red[t] = v;
  __syncthreads();
  for (int off = 64; off > 0; off >>= 1) {
    if (t < off) red[t] = fmaxf(red[t], red[t + off]);
    __syncthreads();
  }
  const float m = red[0];
  __syncthreads();
  const float e = expf(v - m);
  red[t] = e;
  __syncthreads();
  for (int off = 64; off > 0; off >>= 1) {
    if (t < off) red[t] += red[t + off];
    __syncthreads();
  }
  row[t] = e / red[0];
}

// torch permute(0,2,1).view(B,Ci,H,W)  ==  per-batch transpose of Y[128][4096]
__global__ __launch_bounds__(256) void transpose_ci_n(const float* __restrict__ Y,
                                                      float* __restrict__ Z) {
  __shared__ float tile[32][33];
  const int b  = blockIdx.z;
  const int n0 = blockIdx.x * 32;
  const int i0 = blockIdx.y * 32;
  const int tx = threadIdx.x & 31;
  const int ty = threadIdx.x >> 5;  // 0..7
  const float* Yb = Y + (size_t)b * ((size_t)CI_ * N_);
  float*       Zb = Z + (size_t)b * ((size_t)CI_ * N_);
#pragma unroll
  for (int j = 0; j < 32; j += 8)
    tile[ty + j][tx] = Yb[(size_t)(i0 + ty + j) * N_ + n0 + tx];
  __syncthreads();
#pragma unroll
  for (int j = 0; j < 32; j += 8)
    Zb[(size_t)(n0 + ty + j) * CI_ + i0 + tx] = tile[tx][ty + j];
}

// InstanceNorm (affine=False) + ReLU, in place; one workgroup per 4096-row
__global__ __launch_bounds__(256) void inorm_relu(float* __restrict__ P) {
  __shared__ float ra[256], rb[256];
  const int t = threadIdx.x;
  float* p = P + (size_t)blockIdx.x * N_;
  float v[16];
  float s = 0.0f, s2 = 0.0f;
#pragma unroll
  for (int j = 0; j < 16; ++j) {
    v[j] = p[t + 256 * j];
    s += v[j]; s2 += v[j] * v[j];
  }
  ra[t] = s; rb[t] = s2;
  __syncthreads();
  for (int off = 128; off > 0; off >>= 1) {
    if (t < off) { ra[t] += ra[t + off]; rb[t] += rb[t + off]; }
    __syncthreads();
  }
  const float mean = ra[0] * (1.0f / N_);
  const float var  = rb[0] * (1.0f / N_) - mean * mean;
  const float rstd = rsqrtf(var + EPS_);
#pragma unroll
  for (int j = 0; j < 16; ++j)
    p[t + 256 * j] = fmaxf((v[j] - mean) * rstd, 0.0f);
}

// Final InstanceNorm + residual add: out = (U - mean)*rstd + x
__global__ __launch_bounds__(256) void inorm_add(const float* __restrict__ U,
                                                 const float* __restrict__ x,
                                                 float* __restrict__ out) {
  __shared__ float ra[256], rb[256];
  const int t = threadIdx.x;
  const size_t base = (size_t)blockIdx.x * N_;
  float v[16];
  float s = 0.0f, s2 = 0.0f;
#pragma unroll
  for (int j = 0; j < 16; ++j) {
    v[j] = U[base + t + 256 * j];
    s += v[j]; s2 += v[j] * v[j];
  }
  ra[t] = s; rb[t] = s2;
  __syncthreads();
  for (int off = 128; off > 0; off >>= 1) {
    if (t < off) { ra[t] += ra[t + off]; rb[t] += rb[t + off]; }
    __syncthreads();
  }
  const float mean = ra[0] * (1.0f / N_);
  const float var  = rb[0] * (1.0f / N_) - mean * mean;
  const float rstd = rsqrtf(var + EPS_);
#pragma unroll
  for (int j = 0; j < 16; ++j) {
    const size_t o = base + t + 256 * j;
    out[o] = (v[j] - mean) * rstd + x[o];
  }
}

extern "C" void kernel_launch(void* const* d_in, const int* in_sizes, int n_in,
                              void* d_out, int out_size, void* d_ws, size_t ws_size,
                              hipStream_t stream) {
  const float* x    = (const float*)d_in[0];
  const float* g_w  = (const float*)d_in[1];
  const float* g_b  = (const float*)d_in[2];
  const float* th_w = (const float*)d_in[3];
  const float* th_b = (const float*)d_in[4];
  const float* ph_w = (const float*)d_in[5];
  const float* ph_b = (const float*)d_in[6];
  const float* W_w  = (const float*)d_in[7];
  const float* W_b  = (const float*)d_in[8];
  float* out = (float*)d_out;

  float* ws = (float*)d_ws;
  const size_t PN   = (size_t)CI_ * N_;   // 524288 floats per (batch, proj)
  const size_t SLAB = (size_t)B_ * PN;    // 8388608 floats per proj slab
  float* P  = ws;                 // [3][B][Ci][N]   (g, theta, phi)
  float* Y  = P + 3 * SLAB;       // [B][Ci][N]
  float* Z  = Y + SLAB;           // [B][Ci][N]
  float* Fp = Z;                  // alias: split-K partials, dead before Z used
  float* F  = Z + SLAB;           // [B][Ci][Ci]
  float* U  = P;                  // alias: final conv output over dead P

  dim3 blk(256);

  // 1) three 1x1-conv projections (NN GEMM + bias)
  gemm_nn_wmma<<<dim3(32, 1, B_), blk, 0, stream>>>(g_w,  0, x, (long long)C_ * N_, P,            (long long)PN, g_b,  C_, N_);
  gemm_nn_wmma<<<dim3(32, 1, B_), blk, 0, stream>>>(th_w, 0, x, (long long)C_ * N_, P + SLAB,     (long long)PN, th_b, C_, N_);
  gemm_nn_wmma<<<dim3(32, 1, B_), blk, 0, stream>>>(ph_w, 0, x, (long long)C_ * N_, P + 2 * SLAB, (long long)PN, ph_b, C_, N_);
  // 2) instance norm + relu on all 3*B*Ci rows, in place
  inorm_relu<<<dim3(3 * B_ * CI_), blk, 0, stream>>>(P);
  // 3) Gram: theta @ phi^T, split-K=16, then deterministic reduce
  gemm_nt_splitk<<<dim3(16, 1, B_), blk, 0, stream>>>(P + SLAB, P + 2 * SLAB, Fp);
  reduce_split<<<dim3((B_ * CI_ * CI_) / 256), blk, 0, stream>>>(Fp, F);
  // 4) softmax over rows of 128
  softmax_row128<<<dim3(B_ * CI_), dim3(128), 0, stream>>>(F);
  // 5) Y = attn @ g
  gemm_nn_wmma<<<dim3(32, 1, B_), blk, 0, stream>>>(F, (long long)CI_ * CI_, P, (long long)PN, Y, (long long)PN, nullptr, CI_, N_);
  // 6) permute-view == per-batch transpose
  transpose_ci_n<<<dim3(N_ / 32, CI_ / 32, B_), blk, 0, stream>>>(Y, Z);
  // 7) final 1x1 conv (M=256 -> gridDim.y = 2)
  gemm_nn_wmma<<<dim3(32, 2, B_), blk, 0, stream>>>(W_w, 0, Z, (long long)PN, U, (long long)C_ * N_, W_b, CI_, N_);
  // 8) instance norm + residual
  inorm_add<<<dim3(B_ * C_), blk, 0, stream>>>(U, x, out);
}